// Memory_56246891708493
// MI455X (gfx1250) — compile-verified
//
#include <hip/hip_runtime.h>
#include <hip/hip_bf16.h>

#define N_CLASSES 81
#define C_PAD     96
#define BANK      10
#define INS_DIM   2048
#define N_INS     16384
#define KC        128
#define BS_STRIDE 132   // padded row stride (floats): keeps 16B alignment, spreads LDS banks
#define AS_STRIDE 132

typedef __attribute__((ext_vector_type(2))) float v2f;
typedef __attribute__((ext_vector_type(8))) float v8f;

// ---- CDNA5 async Global<->LDS copies (ASYNCcnt-tracked, no VGPR round trip) ----
// saddr form: uniform 64-bit base in SGPR pair + per-lane u32 byte offset in VGPR.
// Generic pointers to __shared__ carry the LDS byte offset in their low 32 bits
// so we pass that as the VDST/VSRC VGPR.
__device__ __forceinline__ void async_load_b128(const void* gbase, unsigned goff, void* lds) {
    asm volatile("global_load_async_to_lds_b128 %0, %1, %2"
                 :
                 : "v"((unsigned)(unsigned long long)(uintptr_t)lds),
                   "v"(goff),
                   "s"((unsigned long long)(uintptr_t)gbase)
                 : "memory");
}
__device__ __forceinline__ void async_store_b128(void* gbase, unsigned goff, const void* lds) {
    asm volatile("global_store_async_from_lds_b128 %0, %1, %2"
                 :
                 : "v"(goff),
                   "v"((unsigned)(unsigned long long)(uintptr_t)lds),
                   "s"((unsigned long long)(uintptr_t)gbase)
                 : "memory");
}
__device__ __forceinline__ void wait_async0() {
    asm volatile("s_wait_asynccnt 0x0" ::: "memory");
}

// ---------------- kernel 1: per-class bank mean (padded to 96 classes) ----------------
__global__ void mean_kernel(const float* __restrict__ memory, float* __restrict__ mem_mean) {
    int c = blockIdx.x;                         // 0..95
    int d = blockIdx.y * 256 + threadIdx.x;     // 0..2047
    float s = 0.0f;
    if (c < N_CLASSES) {
        const float* p = memory + ((size_t)c * BANK) * INS_DIM + d;
        #pragma unroll
        for (int b = 0; b < BANK; ++b) s += p[(size_t)b * INS_DIM];
        s *= (1.0f / (float)BANK);
    }
    mem_mean[(size_t)c * INS_DIM + d] = s;
}

// ---------------- kernel 2: squared norms of mem_mean rows ----------------
__global__ void norm_kernel(const float* __restrict__ mem_mean, float* __restrict__ mem_norm) {
    __shared__ float red[8];
    int c = blockIdx.x;
    int t = threadIdx.x;
    const float* row = mem_mean + (size_t)c * INS_DIM;
    float s = 0.0f;
    for (int d = t; d < INS_DIM; d += 256) { float v = row[d]; s += v * v; }
    for (int off = 16; off; off >>= 1) s += __shfl_down(s, off, 32);
    if ((t & 31) == 0) red[t >> 5] = s;
    __syncthreads();
    if (t == 0) {
        float tot = 0.0f;
        #pragma unroll
        for (int w = 0; w < 8; ++w) tot += red[w];
        mem_norm[c] = (c < N_CLASSES) ? tot : 3.0e38f;   // padding classes never win argmin
    }
}

// ---------------- kernel 3: WMMA GEMM (dot products) + fused argmin ----------------
// block = 256 threads = 8 waves; each wave owns a 16-row strip; block covers 128 rows x 96 classes.
__global__ __launch_bounds__(256) void gemm_argmin_kernel(
    const float* __restrict__ instances,
    const float* __restrict__ mem_mean,
    const float* __restrict__ mem_norm,
    float* __restrict__ cls_out)
{
    extern __shared__ float smem[];
    float* Bs = smem;                               // [C_PAD][BS_STRIDE]
    float* As = smem + C_PAD * BS_STRIDE;           // [128][AS_STRIDE]

    const int tid   = threadIdx.x;
    const int wave  = tid >> 5;
    const int lane  = tid & 31;
    const int laneM = lane & 15;   // M (for A) / N (for B) within tile
    const int laneH = lane >> 4;   // selects K-pair base (0 or 2)
    const int rowBase = blockIdx.x * 128;
    const float* abase = instances + (size_t)rowBase * INS_DIM;

    v8f acc[6];
    #pragma unroll
    for (int t6 = 0; t6 < 6; ++t6) {
        v8f z = {0.0f, 0.0f, 0.0f, 0.0f, 0.0f, 0.0f, 0.0f, 0.0f};
        acc[t6] = z;
    }

    for (int k0 = 0; k0 < INS_DIM; k0 += KC) {
        // stage B chunk: 96 rows x 32 float4, async global->LDS (no VGPR staging)
        for (int idx = tid; idx < C_PAD * (KC / 4); idx += 256) {
            int r = idx >> 5, c4 = idx & 31;
            async_load_b128(mem_mean,
                            (unsigned)((r * INS_DIM + k0 + c4 * 4) * sizeof(float)),
                            Bs + r * BS_STRIDE + c4 * 4);
        }
        // stage A chunk: 128 rows x 32 float4, async global->LDS
        for (int idx = tid; idx < 128 * (KC / 4); idx += 256) {
            int r = idx >> 5, c4 = idx & 31;
            async_load_b128(abase,
                            (unsigned)((r * INS_DIM + k0 + c4 * 4) * sizeof(float)),
                            As + r * AS_STRIDE + c4 * 4);
        }
        if (k0 + KC < INS_DIM)  // hint next A chunk toward the caches
            __builtin_prefetch(abase + (size_t)(tid >> 1) * INS_DIM + k0 + KC, 0, 1);
        wait_async0();
        __syncthreads();

        const float* aRow = As + (wave * 16 + laneM) * AS_STRIDE + laneH * 2;
        const float* bCol = Bs + laneM * BS_STRIDE + laneH * 2;

        #pragma unroll 4
        for (int kk = 0; kk < KC; kk += 4) {
            float2 a2 = *(const float2*)(aRow + kk);
            v2f a = {a2.x, a2.y};
            #pragma unroll
            for (int t6 = 0; t6 < 6; ++t6) {
                float2 b2 = *(const float2*)(bCol + t6 * 16 * BS_STRIDE + kk);
                v2f b = {b2.x, b2.y};
                // D = A(16x4) x B(4x16) + C  -> dot(instances_row, mem_mean_row)
                acc[t6] = __builtin_amdgcn_wmma_f32_16x16x4_f32(
                    false, a, false, b, (short)0, acc[t6], false, false);
            }
        }
        __syncthreads();
    }

    // argmin_n ( mem_norm[n] - 2*dot[m][n] )  — instance norm is constant per row, irrelevant.
    float mn[6];
    #pragma unroll
    for (int t6 = 0; t6 < 6; ++t6) mn[t6] = mem_norm[t6 * 16 + laneM];

    #pragma unroll
    for (int r = 0; r < 8; ++r) {
        float bestV = 3.4e38f;
        int   bestN = 0;
        #pragma unroll
        for (int t6 = 0; t6 < 6; ++t6) {
            int n = t6 * 16 + laneM;
            float v = mn[t6] - 2.0f * acc[t6][r];
            if (v < bestV) { bestV = v; bestN = n; }
        }
        // butterfly min over the 16 lanes of each half-wave (ties -> smallest index)
        #pragma unroll
        for (int m = 1; m < 16; m <<= 1) {
            float ov = __shfl_xor(bestV, m, 32);
            int   oi = __shfl_xor(bestN, m, 32);
            if (ov < bestV || (ov == bestV && oi < bestN)) { bestV = ov; bestN = oi; }
        }
        if (laneM == 0) {
            int row = rowBase + wave * 16 + laneH * 8 + r;   // C/D layout: VGPR r -> M=r (lo), M=8+r (hi)
            cls_out[row] = (float)bestN;
        }
    }
}

// ---------------- kernel 4: accuracy ----------------
__global__ void acc_kernel(const float* __restrict__ cls_out,
                           const int* __restrict__ labels,
                           float* __restrict__ out_acc) {
    __shared__ int red[8];
    int t = threadIdx.x;
    int cnt = 0;
    for (int i = t; i < N_INS; i += 256)
        cnt += ((int)cls_out[i] == labels[i]) ? 1 : 0;
    for (int off = 16; off; off >>= 1) cnt += __shfl_down(cnt, off, 32);
    if ((t & 31) == 0) red[t >> 5] = cnt;
    __syncthreads();
    if (t == 0) {
        int s = 0;
        #pragma unroll
        for (int w = 0; w < 8; ++w) s += red[w];
        *out_acc = (float)s / (float)N_INS;
    }
}

// ---------------- kernel 5: order-dependent bank update, parallel over classes ----------------
__global__ __launch_bounds__(256) void update_kernel(
    const float* __restrict__ instances,
    const int*   __restrict__ labels,
    const float* __restrict__ memory,
    const int*   __restrict__ memory_pos,
    float* __restrict__ new_mem,
    float* __restrict__ new_pos)
{
    extern __shared__ float bank[];           // [BANK * INS_DIM] = 80 KB in LDS
    __shared__ float sred[8 * BANK];
    __shared__ int   sLab[256];
    __shared__ int   sSlot;
    __shared__ int   sPos;

    const int c = blockIdx.x;
    const int t = threadIdx.x;

    // async-stage the whole class bank into LDS (5120 x b128)
    const float* gbank = memory + (size_t)c * BANK * INS_DIM;
    for (int i4 = t; i4 < BANK * INS_DIM / 4; i4 += 256)
        async_load_b128(gbank, (unsigned)(i4 * 16), bank + i4 * 4);
    if (t == 0) sPos = memory_pos[c];
    wait_async0();
    __syncthreads();

    for (int base = 0; base < N_INS; base += 256) {
        sLab[t] = labels[base + t];
        __syncthreads();
        for (int j = 0; j < 256; ++j) {
            if (sLab[j] == c) {                       // block-uniform branch
                const int i = base + j;
                const float* ins = instances + (size_t)i * INS_DIM;
                float insv[INS_DIM / 256];
                #pragma unroll
                for (int e = 0; e < INS_DIM / 256; ++e) insv[e] = ins[t + e * 256];

                int p = sPos;
                __syncthreads();                      // everyone sampled p before any update
                int slot;
                if (p < BANK) {
                    slot = p;
                    if (t == 0) sPos = p + 1;
                } else {
                    // squared distance of instance to each current slot (argmax == farthest)
                    float dist[BANK];
                    #pragma unroll
                    for (int s = 0; s < BANK; ++s) dist[s] = 0.0f;
                    #pragma unroll
                    for (int e = 0; e < INS_DIM / 256; ++e) {
                        float x = insv[e];
                        int d = t + e * 256;
                        #pragma unroll
                        for (int s = 0; s < BANK; ++s) {
                            float diff = bank[s * INS_DIM + d] - x;
                            dist[s] += diff * diff;
                        }
                    }
                    #pragma unroll
                    for (int s = 0; s < BANK; ++s)
                        for (int off = 16; off; off >>= 1)
                            dist[s] += __shfl_down(dist[s], off, 32);
                    if ((t & 31) == 0) {
                        #pragma unroll
                        for (int s = 0; s < BANK; ++s) sred[(t >> 5) * BANK + s] = dist[s];
                    }
                    __syncthreads();
                    if (t == 0) {
                        float best = -1.0f; int bi = 0;
                        #pragma unroll
                        for (int s = 0; s < BANK; ++s) {
                            float tot = 0.0f;
                            #pragma unroll
                            for (int w = 0; w < 8; ++w) tot += sred[w * BANK + s];
                            if (tot > best) { best = tot; bi = s; }   // first max = smallest index
                        }
                        sSlot = bi;
                    }
                    __syncthreads();
                    slot = sSlot;
                }
                // write instance into chosen slot
                #pragma unroll
                for (int e = 0; e < INS_DIM / 256; ++e)
                    bank[slot * INS_DIM + t + e * 256] = insv[e];
                __syncthreads();                      // bank/pos visible before next instance
            }
        }
        __syncthreads();
    }

    // async writeback: LDS -> global, no VGPR staging
    float* gout = new_mem + (size_t)c * BANK * INS_DIM;
    for (int i4 = t; i4 < BANK * INS_DIM / 4; i4 += 256)
        async_store_b128(gout, (unsigned)(i4 * 16), bank + i4 * 4);
    wait_async0();
    if (t == 0) new_pos[c] = (float)sPos;
}

// ---------------- host-side launcher ----------------
extern "C" void kernel_launch(void* const* d_in, const int* in_sizes, int n_in,
                              void* d_out, int out_size, void* d_ws, size_t ws_size,
                              hipStream_t stream) {
    const float* instances  = (const float*)d_in[0];
    const int*   labels     = (const int*)d_in[1];
    const float* memory     = (const float*)d_in[2];
    const int*   memory_pos = (const int*)d_in[3];

    float* out      = (float*)d_out;
    float* cls_out  = out;                                   // [16384]
    float* acc_out  = out + N_INS;                           // [1]
    float* new_mem  = out + N_INS + 1;                       // [81*10*2048]
    float* new_pos  = new_mem + (size_t)N_CLASSES * BANK * INS_DIM; // [81]

    float* mem_mean = (float*)d_ws;                          // [96*2048]
    float* mem_norm = mem_mean + (size_t)C_PAD * INS_DIM;    // [96]

    mean_kernel<<<dim3(C_PAD, INS_DIM / 256), 256, 0, stream>>>(memory, mem_mean);
    norm_kernel<<<C_PAD, 256, 0, stream>>>(mem_mean, mem_norm);

    size_t smemC = (size_t)(C_PAD * BS_STRIDE + 128 * AS_STRIDE) * sizeof(float);
    gemm_argmin_kernel<<<N_INS / 128, 256, smemC, stream>>>(instances, mem_mean, mem_norm, cls_out);

    acc_kernel<<<1, 256, 0, stream>>>(cls_out, labels, acc_out);

    update_kernel<<<N_CLASSES, 256, (size_t)BANK * INS_DIM * sizeof(float), stream>>>(
        instances, labels, memory, memory_pos, new_mem, new_pos);
}